// TokenChoiceRouter_12567074308597
// MI455X (gfx1250) — compile-verified
//
#include <hip/hip_runtime.h>
#include <math.h>

// Problem constants from the reference
#define BB 8
#define TT 8192
#define DD 512
#define NTOK (BB * TT)           // 65536 tokens
#define TOK_PER_WAVE 16
#define NGROUP (NTOK / TOK_PER_WAVE)  // 4096 waves of work

typedef __attribute__((ext_vector_type(2))) float v2f;
typedef __attribute__((ext_vector_type(8))) float v8f;

static __device__ __forceinline__ float sigmoidf_dev(float v) {
    return 1.0f / (1.0f + expf(-v));
}

// -------------------------------------------------------------------------
// Kernel 1: GEMV logits[tok] = dot(x[tok,:], W[:]) via V_WMMA_F32_16X16X4_F32.
// A-matrix (16x4): row 0 = W[k..k+3] (lanes 0 and 16 per ISA layout; other
// rows are don't-care since we only read D row 0). Fed from LDS (broadcast).
// B-matrix (4x16): column n = token n. Per ISA 32-bit layout, lane n (<16)
// holds (K=0,K=1) and lane n+16 holds (K=2,K=3) -> one contiguous b64 load
// per lane per step, streaming each token row sequentially. Result: D row 0
// = acc[0] lanes 0..15 = the 16 token logits.
// -------------------------------------------------------------------------
__global__ __launch_bounds__(256) void router_gemv_wmma(
    const float* __restrict__ x, const float* __restrict__ W,
    float* __restrict__ logits) {
    __shared__ float wlds[DD];
    const int tid = threadIdx.x;
    // stage W into LDS once per block (512 floats, 256 threads)
    wlds[tid]       = W[tid];
    wlds[tid + 256] = W[tid + 256];
    __syncthreads();

    const int lane  = tid & 31;
    const int wave  = tid >> 5;
    const int group = blockIdx.x * 8 + wave;   // grid sized exactly: no guard,
                                               // EXEC stays all-ones for WMMA
    const int tok   = group * TOK_PER_WAVE + (lane & 15);
    // lanes >=16 read K+2,K+3 of their token's row
    const float* xp = x + (size_t)tok * DD + ((lane >> 4) << 1);
    const int wofs  = (lane >> 4) << 1;        // 0 for lanes<16, 2 for >=16

    v8f acc = {};
#pragma unroll 4
    for (int k = 0; k < DD; k += 4) {
        v2f a = *(const v2f*)(&wlds[k + wofs]);      // ds_load_b64 (broadcast)
        v2f b = *(const v2f*)(xp + k);               // global_load_b64, coalesced stream
        acc = __builtin_amdgcn_wmma_f32_16x16x4_f32(
            /*neg_a=*/false, a, /*neg_b=*/false, b,
            /*c_mod=*/(short)0, acc, /*reuse_a=*/false, /*reuse_b=*/false);
    }
    if (lane < 16) {
        logits[group * TOK_PER_WAVE + lane] = acc[0];  // D row 0, N=lane
    }
}

// -------------------------------------------------------------------------
// Kernel 2: per-batch stats. One block per b.
//   ws[b]     = z[b] = logsumexp_t(logits[b,:])
//   ws[8 + b] = sum_t sigmoid(logits[b,t])
// Deterministic tree reductions (no atomics).
// -------------------------------------------------------------------------
__global__ __launch_bounds__(256) void router_stats(
    const float* __restrict__ logits, float* __restrict__ ws) {
    const int b = blockIdx.x;
    const float* lb = logits + b * TT;
    __shared__ float smax[256];
    __shared__ float ssum[256];
    const int tid = threadIdx.x;

    float mx = -3.402823466e38f;
    float sg = 0.0f;
    for (int i = tid; i < TT; i += 256) {
        float v = lb[i];
        mx = fmaxf(mx, v);
        sg += sigmoidf_dev(v);
    }
    smax[tid] = mx;
    ssum[tid] = sg;
    __syncthreads();
    for (int s = 128; s > 0; s >>= 1) {
        if (tid < s) {
            smax[tid] = fmaxf(smax[tid], smax[tid + s]);
            ssum[tid] += ssum[tid + s];
        }
        __syncthreads();
    }
    const float maxl  = smax[0];
    const float sgtot = ssum[0];
    __syncthreads();

    float es = 0.0f;
    for (int i = tid; i < TT; i += 256) es += expf(lb[i] - maxl);
    smax[tid] = es;
    __syncthreads();
    for (int s = 128; s > 0; s >>= 1) {
        if (tid < s) smax[tid] += smax[tid + s];
        __syncthreads();
    }
    if (tid == 0) {
        ws[b]     = maxl + logf(smax[0]);
        ws[8 + b] = sgtot;
    }
}

// -------------------------------------------------------------------------
// Kernel 3: scalar losses. scal[0] = aux_loss, scal[1] = z_loss.
// -------------------------------------------------------------------------
__global__ void router_scalars(const float* __restrict__ ws,
                               float* __restrict__ scal) {
    float sg = 0.0f, zz = 0.0f;
    for (int b = 0; b < BB; ++b) {
        sg += ws[8 + b];
        float z = ws[b];
        zz += z * z;
    }
    float m = sg / (float)NTOK;
    scal[0] = 0.01f * m * (1.0f - m);          // BAL_W * m * (1 - m)
    scal[1] = 0.001f * (zz / (float)BB);       // Z_W * mean(z^2)
}

// -------------------------------------------------------------------------
// Kernel 4: stable compaction. One wave32 per b. Ballot+popc prefix produces
// the stable-argsort order (selected token indices in increasing t). Tail
// positions j >= count replicate the last selected index (token 0 if none),
// matching idx = order[min(j, count-1)].
// -------------------------------------------------------------------------
__global__ __launch_bounds__(32) void router_compact(
    const float* __restrict__ logits, float* __restrict__ sel,
    float* __restrict__ gw, int max_k) {
    const int b = blockIdx.x;
    const int lane = threadIdx.x;
    const float* lb = logits + b * TT;

    int base = 0;
    int lastIdx = -1;
    for (int c = 0; c < TT; c += 32) {
        float v = lb[c + lane];
        bool pred = (v >= 0.0f);               // sigmoid(v) >= 0.5  <=>  v >= 0
        unsigned m = (unsigned)__ballot(pred); // wave32: low 32 bits
        int pos = base + __popc(m & ((1u << lane) - 1u));
        if (pred && pos < max_k) {
            sel[b * max_k + pos] = (float)(c + lane);
            gw[b * max_k + pos]  = sigmoidf_dev(v);
        }
        base += __popc(m);
        if (m) lastIdx = c + 31 - __clz((int)m);
    }

    int fillIdx;
    if (base == 0) fillIdx = 0;                 // no selected: order[0] == token 0
    else           fillIdx = lastIdx;
    float fillG = sigmoidf_dev(lb[fillIdx]);
    int start = (base < max_k) ? base : max_k;
    for (int j = start + lane; j < max_k; j += 32) {
        sel[b * max_k + j] = (float)fillIdx;
        gw[b * max_k + j]  = fillG;
    }
}

extern "C" void kernel_launch(void* const* d_in, const int* in_sizes, int n_in,
                              void* d_out, int out_size, void* d_ws, size_t ws_size,
                              hipStream_t stream) {
    const float* x = (const float*)d_in[0];   // (8, 8192, 512) f32
    const float* W = (const float*)d_in[1];   // (1, 512) f32
    float* out = (float*)d_out;
    float* ws  = (float*)d_ws;

    // out = [selected(B*max_k) | gate_weights(B*max_k) | raw_logits(B*T) | aux | z]
    int max_k = (out_size - (BB * TT + 2)) / (2 * BB);
    if (max_k < 1) max_k = 1;
    float* sel    = out;
    float* gw     = out + (size_t)BB * max_k;
    float* logits = out + (size_t)2 * BB * max_k;
    float* scal   = logits + (size_t)BB * TT;

    router_gemv_wmma<<<NGROUP / 8, 256, 0, stream>>>(x, W, logits);
    router_stats<<<BB, 256, 0, stream>>>(logits, ws);
    router_scalars<<<1, 1, 0, stream>>>(ws, scal);
    router_compact<<<BB, 32, 0, stream>>>(logits, sel, gw, max_k);
}